// _SparseAttentionLayer_34849364640427
// MI455X (gfx1250) — compile-verified
//
#include <hip/hip_runtime.h>
#include <math.h>

#define NODES   50000
#define EDGES   800000
#define HID     128
#define EDIM    16
#define HEADS   8
#define HDIM    16
#define QKSCALE 0.25f            /* 1/sqrt(16) */
#define LN_EPS  1e-5f

typedef float v2f __attribute__((ext_vector_type(2)));
typedef float v8f __attribute__((ext_vector_type(8)));

// ---------------------------------------------------------------- utilities
__global__ void fill_kernel(float* __restrict__ p, float val, int n) {
    int i = blockIdx.x * blockDim.x + threadIdx.x;
    if (i < n) p[i] = val;
}

__device__ __forceinline__ void atomicMaxF(float* addr, float value) {
    if (__float_as_int(value) >= 0) {
        atomicMax((int*)addr, __float_as_int(value));
    } else {
        atomicMin((unsigned int*)addr, __float_as_uint(value));
    }
}

__device__ __forceinline__ float siluf(float x) {
    return x * (1.0f / (1.0f + expf(-x)));
}

// ------------------------------------------------- WMMA f32 GEMM (+bias,+act)
// out[M,N] = act(A[M,K] @ W[K,N] + bias[N]);  act: 0 = none, 1 = silu
// grid.x = M/16,   blockDim.x = 32 * (N/16)   (one wave per 16x16 output tile)
// A tile (16xK) staged in LDS; W streamed from global (L2-resident, shared).
// f32 WMMA lane layout (ISA 7.12.2):
//   A 16x4 : row = lane&15 ; VGPR0/1 = K+0/K+1 (lanes<16) or K+2/K+3 (lanes>=16)
//   B 4x16 : col = lane&15 ; same K-pair split by lane>>4
//   C/D    : VGPR j -> row j (lanes<16) / row j+8 (lanes>=16), col = lane&15
__global__ void gemm_wmma_f32(const float* __restrict__ A,
                              const float* __restrict__ W,
                              const float* __restrict__ bias,
                              float* __restrict__ out,
                              int M, int K, int N, int act)
{
    __shared__ float sA[16 * 256];          // supports K up to 256 (16 KB)
    const int tid     = threadIdx.x;
    const int nthr    = blockDim.x;
    const int mbase   = blockIdx.x * 16;

    for (int i = tid; i < 16 * K; i += nthr) {
        int r = i / K, c = i - r * K;
        sA[i] = A[(size_t)(mbase + r) * K + c];
    }
    __syncthreads();

    const int wave = tid >> 5;
    const int lane = tid & 31;
    const int col  = wave * 16 + (lane & 15);
    const int kb   = (lane >> 4) * 2;       // K sub-offset for this half-wave
    const int arow = lane & 15;

    v8f acc = {};
    {
        const float b0 = bias[col];
        #pragma unroll
        for (int j = 0; j < 8; ++j) acc[j] = b0;
    }

    #pragma unroll 8
    for (int k = 0; k < K; k += 4) {
        v2f a, b;
        a.x = sA[arow * K + k + kb];
        a.y = sA[arow * K + k + kb + 1];
        b.x = W[(size_t)(k + kb)     * N + col];
        b.y = W[(size_t)(k + kb + 1) * N + col];
        acc = __builtin_amdgcn_wmma_f32_16x16x4_f32(
                  false, a, false, b, (short)0, acc, false, false);
    }

    const int rowhi = (lane >> 4) * 8;
    #pragma unroll
    for (int j = 0; j < 8; ++j) {
        float val = acc[j];
        if (act == 1) val = siluf(val);
        out[(size_t)(mbase + j + rowhi) * N + col] = val;
    }
}

// --------------------------------------------- per-(edge,head) logits + bias
__global__ void edge_logits_kernel(const float* __restrict__ q,
                                   const float* __restrict__ k,
                                   const int*   __restrict__ ei,
                                   const float* __restrict__ ef,
                                   const float* __restrict__ Wb1,
                                   const float* __restrict__ bb1,
                                   const float* __restrict__ Wb2,
                                   const float* __restrict__ bb2,
                                   float* __restrict__ logits,
                                   float* __restrict__ maxbuf)
{
    long tid = (long)blockIdx.x * blockDim.x + threadIdx.x;
    if (tid >= (long)EDGES * HEADS) return;
    const int e = (int)(tid >> 3);
    const int h = (int)(tid & 7);
    const int src = ei[e];
    const int dst = ei[EDGES + e];

    // edge bias MLP: silu(ef @ Wb1 + bb1) @ Wb2 + bb2  (column h only)
    float bias = bb2[h];
    #pragma unroll
    for (int j = 0; j < HEADS; ++j) {
        float t = bb1[j];
        #pragma unroll
        for (int i = 0; i < EDIM; ++i)
            t += ef[(size_t)e * EDIM + i] * Wb1[i * HEADS + j];
        bias += siluf(t) * Wb2[j * HEADS + h];
    }

    const float* qp = q + (size_t)dst * HID + h * HDIM;
    const float* kp = k + (size_t)src * HID + h * HDIM;
    float dot = 0.f;
    #pragma unroll
    for (int d = 0; d < HDIM; ++d) dot += qp[d] * kp[d];

    const float lg = dot * QKSCALE + bias;
    logits[tid] = lg;
    atomicMaxF(&maxbuf[(size_t)dst * HEADS + h], lg);
}

// --------------------------------------------- exp(logit - max) + segment sum
__global__ void edge_exp_kernel(const int* __restrict__ ei,
                                float* __restrict__ logits,   // in: logit, out: exp
                                const float* __restrict__ maxbuf,
                                float* __restrict__ norm)
{
    long tid = (long)blockIdx.x * blockDim.x + threadIdx.x;
    if (tid >= (long)EDGES * HEADS) return;
    const int e = (int)(tid >> 3);
    const int h = (int)(tid & 7);
    const int dst = ei[EDGES + e];
    const float ex = expf(logits[tid] - maxbuf[(size_t)dst * HEADS + h]);
    logits[tid] = ex;
    atomicAdd(&norm[(size_t)dst * HEADS + h], ex);
}

// --------------------------------- weighted scatter-add: agg[dst] += attn*v[src]
__global__ void aggregate_kernel(const int* __restrict__ ei,
                                 const float* __restrict__ expw,
                                 const float* __restrict__ norm,
                                 const float* __restrict__ v,
                                 float* __restrict__ agg)
{
    long tid = (long)blockIdx.x * blockDim.x + threadIdx.x;
    if (tid >= (long)EDGES * HID) return;
    const int e = (int)(tid >> 7);
    const int c = (int)(tid & 127);
    const int h = c >> 4;
    const int src = ei[e];
    const int dst = ei[EDGES + e];
    const float w = expw[(size_t)e * HEADS + h]
                    / fmaxf(norm[(size_t)dst * HEADS + h], 1e-12f);
    atomicAdd(&agg[(size_t)dst * HID + c], w * v[(size_t)src * HID + c]);
}

// ------------------------------------------ out = LN(x + y) * g + b, wave/row
__global__ void ln_residual_kernel(const float* __restrict__ x,
                                   const float* __restrict__ y,
                                   const float* __restrict__ g,
                                   const float* __restrict__ b,
                                   float* __restrict__ out, int M)
{
    const int wave = threadIdx.x >> 5;
    const int lane = threadIdx.x & 31;
    const int row  = blockIdx.x * 8 + wave;       // blockDim = 256 -> 8 waves
    if (row >= M) return;

    const float* xp = x + (size_t)row * HID;
    const float* yp = y + (size_t)row * HID;
    float vals[4], s = 0.f, ss = 0.f;
    #pragma unroll
    for (int i = 0; i < 4; ++i) {
        int c = lane * 4 + i;
        float t = xp[c] + yp[c];
        vals[i] = t; s += t; ss += t * t;
    }
    #pragma unroll
    for (int m = 16; m >= 1; m >>= 1) {           // wave32 reduction
        s  += __shfl_xor(s,  m, 32);
        ss += __shfl_xor(ss, m, 32);
    }
    const float mu  = s * (1.0f / HID);
    const float var = ss * (1.0f / HID) - mu * mu;
    const float inv = rsqrtf(var + LN_EPS);
    #pragma unroll
    for (int i = 0; i < 4; ++i) {
        int c = lane * 4 + i;
        out[(size_t)row * HID + c] = (vals[i] - mu) * inv * g[c] + b[c];
    }
}

// --------------------------------------------------------------------- driver
extern "C" void kernel_launch(void* const* d_in, const int* in_sizes, int n_in,
                              void* d_out, int out_size, void* d_ws, size_t ws_size,
                              hipStream_t stream)
{
    const float* nf  = (const float*)d_in[0];
    const int*   ei  = (const int*)  d_in[1];
    const float* ef  = (const float*)d_in[2];
    const float* Wq  = (const float*)d_in[3];  const float* bq  = (const float*)d_in[4];
    const float* Wk  = (const float*)d_in[5];  const float* bk  = (const float*)d_in[6];
    const float* Wv  = (const float*)d_in[7];  const float* bv  = (const float*)d_in[8];
    const float* Wb1 = (const float*)d_in[9];  const float* bb1 = (const float*)d_in[10];
    const float* Wb2 = (const float*)d_in[11]; const float* bb2 = (const float*)d_in[12];
    const float* Wo  = (const float*)d_in[13]; const float* bo  = (const float*)d_in[14];
    const float* Wf1 = (const float*)d_in[15]; const float* bf1 = (const float*)d_in[16];
    const float* Wf2 = (const float*)d_in[17]; const float* bf2 = (const float*)d_in[18];
    const float* g1  = (const float*)d_in[19]; const float* be1 = (const float*)d_in[20];
    const float* g2  = (const float*)d_in[21]; const float* be2 = (const float*)d_in[22];

    // workspace layout (floats), aliasing by lifetime:
    const size_t NH  = (size_t)NODES * HID;     // 6.4M
    const size_t E8  = (size_t)EDGES * HEADS;   // 6.4M
    const size_t NHh = (size_t)NODES * HEADS;   // 0.4M
    float* ws   = (float*)d_ws;
    float* q    = ws;                 // [NH]   -> later reused as FFN hidden (with k)
    float* kk   = q    + NH;          // [NH]
    float* v    = kk   + NH;          // [NH]   -> later reused as FFN output y
    float* attn = v    + NH;          // [E8]   logits -> exp weights; later Wo-out tmp
    float* maxb = attn + E8;          // [NHh]
    float* norm = maxb + NHh;         // [NHh]
    float* agg  = norm + NHh;         // [NH]
    float* x1   = agg  + NH;          // [NH]
    float* hffn = q;                  // [NODES*256] overlays q+kk (both dead by then)
    float* yffn = v;                  // [NH]       overlays v    (dead by then)
    float* tmpo = attn;               // [NH]       overlays attn (dead by then)
    float* outp = (float*)d_out;

    const int MT = NODES / 16;        // 3125 row tiles

    // deterministic per-call init
    fill_kernel<<<(int)((NHh + 255) / 256), 256, 0, stream>>>(maxb, -INFINITY, (int)NHh);
    fill_kernel<<<(int)((NHh + 255) / 256), 256, 0, stream>>>(norm, 0.0f, (int)NHh);
    fill_kernel<<<(int)((NH  + 255) / 256), 256, 0, stream>>>(agg,  0.0f, (int)NH);

    // Q, K, V projections (WMMA f32)
    gemm_wmma_f32<<<MT, 256, 0, stream>>>(nf, Wq, bq, q,  NODES, HID, HID, 0);
    gemm_wmma_f32<<<MT, 256, 0, stream>>>(nf, Wk, bk, kk, NODES, HID, HID, 0);
    gemm_wmma_f32<<<MT, 256, 0, stream>>>(nf, Wv, bv, v,  NODES, HID, HID, 0);

    // edge logits + segment max
    const int EHB = (int)((E8 + 255) / 256);
    edge_logits_kernel<<<EHB, 256, 0, stream>>>(q, kk, ei, ef, Wb1, bb1, Wb2, bb2,
                                                attn, maxb);
    // exp + segment sum
    edge_exp_kernel<<<EHB, 256, 0, stream>>>(ei, attn, maxb, norm);

    // weighted aggregation (scatter-add)
    const long EC = (long)EDGES * HID;
    aggregate_kernel<<<(int)((EC + 255) / 256), 256, 0, stream>>>(ei, attn, norm, v, agg);

    // x1 = LN(node + agg @ Wo + bo)
    gemm_wmma_f32<<<MT, 256, 0, stream>>>(agg, Wo, bo, tmpo, NODES, HID, HID, 0);
    ln_residual_kernel<<<(NODES + 7) / 8, 256, 0, stream>>>(nf, tmpo, g1, be1, x1, NODES);

    // FFN: h = silu(x1 @ Wf1 + bf1); y = h @ Wf2 + bf2; out = LN(x1 + y)
    gemm_wmma_f32<<<MT, 512, 0, stream>>>(x1,   Wf1, bf1, hffn, NODES, HID,     2 * HID, 1);
    gemm_wmma_f32<<<MT, 256, 0, stream>>>(hffn, Wf2, bf2, yffn, NODES, 2 * HID, HID,     0);
    ln_residual_kernel<<<(NODES + 7) / 8, 256, 0, stream>>>(x1, yffn, g2, be2, outp, NODES);
}